// CollisionRegularizer_44727789420690
// MI455X (gfx1250) — compile-verified
//
#include <hip/hip_runtime.h>
#include <hip/hip_bf16.h>

// CDNA5 / gfx1250: wave32, WMMA 16x16x4 f32 for the Gram-matrix tiles.

typedef __attribute__((ext_vector_type(2))) float v2f;
typedef __attribute__((ext_vector_type(8))) float v8f;

#define COLL_RADIUS 0.1f
#define NPTS 8192
#define NTILES (NPTS / 16)   // 512 tiles of 16 rows/cols
#define WAVES_PER_BLOCK 8
#define POISON 1.0e10f       // forces t = max(R - sqrt(~1e10), 0) = 0

// One block owns one (batch, row-tile). All waves stride over the strictly
// upper-triangular tiles (jt > rb, counted twice by symmetry) with a
// BLOCK-UNIFORM trip count so the loop is scalar and EXEC stays all-ones
// around every WMMA (ISA 7.12 requirement). Overshooting waves and the
// non-wave-0 copies of the diagonal tile are neutralized by poisoning the
// column sq-norm, which zeroes their contribution with one cndmask per tile.
__global__ __launch_bounds__(256)
void CollisionRegularizer_44727789420690_tiles(const float* __restrict__ xyz,
                                               float* __restrict__ partial)
{
    const int lane = threadIdx.x & 31;
    const int wave = threadIdx.x >> 5;
    const int rb   = blockIdx.x % NTILES;   // row tile index
    const int b    = blockIdx.x / NTILES;   // batch index
    const float* __restrict__ P = xyz + (size_t)b * NPTS * 3;

    const int half = lane >> 4;             // 0: lanes 0-15, 1: lanes 16-31
    const int lid  = lane & 15;

    // ---- Row-block data (loop invariant) -------------------------------
    const int r0    = rb * 16;
    const int rowPt = r0 + lid;             // lanes l and l+16 load same point
    const float rx = P[rowPt * 3 + 0];
    const float ry = P[rowPt * 3 + 1];
    const float rz = P[rowPt * 3 + 2];
    const float sqa = rx * rx + ry * ry + rz * rz;

    // A fragment, 32-bit A-matrix 16x4 layout (ISA 7.12.2):
    //   lanes 0-15 : M=lid, VGPR0=K0, VGPR1=K1  -> (x, y)
    //   lanes 16-31: M=lid, VGPR0=K2, VGPR1=K3  -> (z, 0)
    v2f afrag;
    afrag.x = half ? rz : rx;
    afrag.y = half ? 0.0f : ry;

    // |p|^2 for the 8 rows this lane-half owns in the D layout:
    // D VGPR r, lanes 0-15 -> row r; lanes 16-31 -> row r+8.
    float rowq[8];
#pragma unroll
    for (int r = 0; r < 8; ++r)
        rowq[r] = __shfl(sqa, r + 8 * half, 32);

    float acc2 = 0.0f;   // sum over off-diagonal tiles (weighted x2 at end)

    // Block-uniform iteration count -> scalar loop, EXEC untouched.
    const int nOff  = NTILES - 1 - rb;
    const int iters = (nOff + WAVES_PER_BLOCK - 1) / WAVES_PER_BLOCK;

    int jt = rb + 1 + wave;
    for (int i = 0; i < iters; ++i, jt += WAVES_PER_BLOCK) {
        const bool valid = (jt < NTILES);
        const int colPt = (valid ? jt * 16 : 0) + lid;   // in-bounds always
        const float bx = P[colPt * 3 + 0];
        const float by = P[colPt * 3 + 1];
        const float bz = P[colPt * 3 + 2];
        float sqb = bx * bx + by * by + bz * bz;         // |p_col|^2, per lane
        if (!valid) sqb = POISON;                        // zero contribution

        // B fragment, 4x16: K rows striped across lane halves (mirrors A).
        v2f bfrag;
        bfrag.x = half ? bz : bx;
        bfrag.y = half ? 0.0f : by;

        v8f c = {};
        c = __builtin_amdgcn_wmma_f32_16x16x4_f32(
            false, afrag, false, bfrag, (short)0, c, false, false);

#pragma unroll
        for (int r = 0; r < 8; ++r) {
            float sq = fmaxf(rowq[r] + sqb - 2.0f * c[r], 0.0f);
            float t  = fmaxf(COLL_RADIUS - __builtin_amdgcn_sqrtf(sq), 0.0f);
            acc2 = fmaf(t, t, acc2);
        }
    }

    float acc = 2.0f * acc2;

    // ---- Diagonal tile (jt == rb): computed by ALL waves (uniform flow,
    // EXEC all-ones for the WMMA); waves != 0 are poisoned to contribute 0.
    {
        const float sqb = (wave == 0) ? sqa : POISON;
        v8f c = {};
        c = __builtin_amdgcn_wmma_f32_16x16x4_f32(
            false, afrag, false, afrag, (short)0, c, false, false);
#pragma unroll
        for (int r = 0; r < 8; ++r) {
            float sq = fmaxf(rowq[r] + sqb - 2.0f * c[r], 0.0f);
            float t  = fmaxf(COLL_RADIUS - __builtin_amdgcn_sqrtf(sq), 0.0f);
            float v  = t * t;
            if (lid == r + 8 * half) v = 0.0f;   // mask diagonal element
            acc += v;
        }
    }

    // ---- Deterministic block reduction ---------------------------------
#pragma unroll
    for (int off = 16; off > 0; off >>= 1)
        acc += __shfl_down(acc, off, 32);

    __shared__ float smem[WAVES_PER_BLOCK];
    if (lane == 0) smem[wave] = acc;
    __syncthreads();
    if (threadIdx.x == 0) {
        float s = 0.0f;
#pragma unroll
        for (int w2 = 0; w2 < WAVES_PER_BLOCK; ++w2) s += smem[w2];
        partial[blockIdx.x] = s;
    }
}

// Single-block fold of the per-block partials; applies the mean scale.
__global__ __launch_bounds__(256)
void CollisionRegularizer_44727789420690_reduce(const float* __restrict__ partial,
                                                float* __restrict__ out,
                                                int n, float scale)
{
    __shared__ float sm[256];
    float s = 0.0f;
    for (int i = threadIdx.x; i < n; i += 256) s += partial[i];
    sm[threadIdx.x] = s;
    __syncthreads();
    for (int stride = 128; stride > 0; stride >>= 1) {
        if (threadIdx.x < stride) sm[threadIdx.x] += sm[threadIdx.x + stride];
        __syncthreads();
    }
    if (threadIdx.x == 0) out[0] = sm[0] * scale;
}

extern "C" void kernel_launch(void* const* d_in, const int* in_sizes, int n_in,
                              void* d_out, int out_size, void* d_ws, size_t ws_size,
                              hipStream_t stream) {
    const float* xyz = (const float*)d_in[0];
    float* out       = (float*)d_out;
    float* partial   = (float*)d_ws;

    const int B = in_sizes[0] / (NPTS * 3);        // 2
    const int nblocks = B * NTILES;                // 1024

    CollisionRegularizer_44727789420690_tiles<<<nblocks, 256, 0, stream>>>(xyz, partial);

    const float scale = 1.0f / ((float)B * (float)NPTS * (float)NPTS); // 2^-27 exactly
    CollisionRegularizer_44727789420690_reduce<<<1, 256, 0, stream>>>(partial, out, nblocks, scale);
}